// RNN_No_FFNN_39917426049311
// MI455X (gfx1250) — compile-verified
//
#include <hip/hip_runtime.h>
#include <cstdint>

#define H       4096
#define DD      104
#define KSPLIT  8
#define TILE_M  16
#define WAVES   4
#define BK      (H / KSPLIT)        // 512 K-cols per block
#define WK      (BK / WAVES)        // 128 K-cols per wave
#define SCOLS   64                  // K-cols per stage
#define NSTAGE  (WK / SCOLS)        // 2
#define LSTRIDE 68                  // LDS row stride (floats): 4r banks, conflict-free
#define TILE_FLOATS (TILE_M * LSTRIDE)

#define USE_ASYNC 1

typedef float v2f __attribute__((ext_vector_type(2)));
typedef float v8f __attribute__((ext_vector_type(8)));

#if USE_ASYNC
#define STAGE_WAIT_NEXT() asm volatile("s_wait_asynccnt 8" ::: "memory")
#define STAGE_WAIT_ALL()  asm volatile("s_wait_asynccnt 0" ::: "memory")
#else
#define STAGE_WAIT_NEXT() __syncthreads()
#define STAGE_WAIT_ALL()  __syncthreads()
#endif

// Stage a 16-row x 64-col fp32 tile of Uw into this wave's private LDS buffer.
// 8 x b128 per lane; rows stored with LSTRIDE padding for bank-conflict-free
// WMMA-layout reads. Async path: GVS mode (saddr = Uw base, vaddr = u32 offset).
__device__ __forceinline__ void stage_load(const float* __restrict__ Uw, int rb, int c0,
                                           float* buf, int lane) {
#if USE_ASYNC
  const uint64_t base = (uint64_t)(uintptr_t)Uw;
#pragma unroll
  for (int i = 0; i < 8; ++i) {
    const int f   = i * 32 + lane;
    const int row = f >> 4;
    const int q   = f & 15;
    const uint32_t goff = ((uint32_t)(rb + row) * (uint32_t)H + (uint32_t)(c0 + q * 4)) * 4u;
    const uint32_t loff = (uint32_t)(uintptr_t)(buf + row * LSTRIDE + q * 4);
    asm volatile("global_load_async_to_lds_b128 %0, %1, %2"
                 :: "v"(loff), "v"(goff), "s"(base)
                 : "memory");
  }
#else
#pragma unroll
  for (int i = 0; i < 8; ++i) {
    const int f   = i * 32 + lane;
    const int row = f >> 4;
    const int q   = f & 15;
    const float4 v = *(const float4*)(Uw + (size_t)(rb + row) * H + (c0 + q * 4));
    *(float4*)(buf + row * LSTRIDE + q * 4) = v;
  }
#endif
}

// y_part[ks][rb+m] = sum_{k in block strip} Uw[rb+m][k] * hidden[k]  (+ Ww@line if ks==0)
__global__ void __launch_bounds__(128)
k_gemv_wmma(const float* __restrict__ Uw, const float* __restrict__ hidden,
            const float* __restrict__ Ww, const float* __restrict__ line,
            float* __restrict__ part) {
  __shared__ __align__(16) float sh_tiles[WAVES * 2 * TILE_FLOATS];
  __shared__ __align__(16) float sh_hid[BK];
  __shared__ float sh_red[WAVES * TILE_M];
  __shared__ float sh_ww[TILE_M];

  const int tid  = threadIdx.x;
  const int wave = tid >> 5;
  const int lane = tid & 31;
  const int rb   = blockIdx.x * TILE_M;   // row tile base
  const int ks   = blockIdx.y;            // K-split index
  const int bk   = ks * BK;               // block K base

  // hidden slice for this block's K strip
  for (int i = tid; i < BK; i += 128) sh_hid[i] = hidden[bk + i];

  float* wbuf0 = &sh_tiles[(wave * 2 + 0) * TILE_FLOATS];
  float* wbuf1 = &sh_tiles[(wave * 2 + 1) * TILE_FLOATS];
  const int wk0 = bk + wave * WK;

  // prefetch stage 0 (per-wave private buffer, no barrier needed for async path)
  stage_load(Uw, rb, wk0, wbuf0, lane);

  // small Ww @ line contribution, only once per row tile (ks == 0)
  if (ks == 0) {
#pragma unroll
    for (int rr = 0; rr < TILE_M / WAVES; ++rr) {
      const int row = rb + wave * (TILE_M / WAVES) + rr;
      float v = 0.0f;
      for (int idx = lane; idx < DD; idx += 32) v += Ww[row * DD + idx] * line[idx];
      for (int o = 16; o > 0; o >>= 1) v += __shfl_down(v, o);
      if (lane == 0) sh_ww[wave * (TILE_M / WAVES) + rr] = v;
    }
  }

  __syncthreads();  // sh_hid (and sh_ww) visible

  v8f acc = {};
  const int rsel = lane & 15;         // A row (M)
  const int ksel = (lane >> 4) * 2;   // K sub-slot: lanes 0-15 -> K 0,1 ; lanes 16-31 -> K 2,3

  for (int s = 0; s < NSTAGE; ++s) {
    float* cur = (s & 1) ? wbuf1 : wbuf0;
    if (s + 1 < NSTAGE) {
      float* nxt = (s & 1) ? wbuf0 : wbuf1;
      stage_load(Uw, rb, wk0 + (s + 1) * SCOLS, nxt, lane);
      STAGE_WAIT_NEXT();   // <=8 outstanding -> current stage complete
    } else {
      STAGE_WAIT_ALL();
    }
    const int hb = wave * WK + s * SCOLS;
#pragma unroll
    for (int j = 0; j < SCOLS / 4; ++j) {
      // A: 16x4 tile of Uw from LDS (ds_load_b64, conflict-free with LSTRIDE=68)
      v2f a = *(const v2f*)&cur[rsel * LSTRIDE + 4 * j + ksel];
      // B: hidden[k..k+4) replicated across all 16 N columns
      v2f b = *(const v2f*)&sh_hid[hb + 4 * j + ksel];
      acc = __builtin_amdgcn_wmma_f32_16x16x4_f32(false, a, false, b,
                                                  (short)0, acc, false, false);
    }
  }

  // C/D layout: VGPR j, lanes 0-15 -> M=j (N=lane); lanes 16-31 -> M=8+j.
  // All N columns identical, so lane 0 / lane 16 hold the 16 row partials.
  if (lane == 0) {
#pragma unroll
    for (int j = 0; j < 8; ++j) sh_red[wave * TILE_M + j] = acc[j];
  }
  if (lane == 16) {
#pragma unroll
    for (int j = 0; j < 8; ++j) sh_red[wave * TILE_M + 8 + j] = acc[j];
  }
  __syncthreads();

  if (tid < TILE_M) {
    float s = sh_red[tid] + sh_red[TILE_M + tid] + sh_red[2 * TILE_M + tid] + sh_red[3 * TILE_M + tid];
    if (ks == 0) s += sh_ww[tid];
    part[ks * H + rb + tid] = s;
  }
}

// h_t = sigmoid(sum_ks part + Ub + Wb); write to d_out[DD + r]
__global__ void k_finish_h(const float* __restrict__ part, const float* __restrict__ Ub,
                           const float* __restrict__ Wb, float* __restrict__ out) {
  const int r = blockIdx.x * blockDim.x + threadIdx.x;
  float acc = Ub[r] + Wb[r];
#pragma unroll
  for (int s = 0; s < KSPLIT; ++s) acc += part[s * H + r];
  out[DD + r] = 1.0f / (1.0f + expf(-acc));
}

// pred[i] = Vw[i,:] @ h_t + Vb[i]
__global__ void k_pred(const float* __restrict__ Vw, const float* __restrict__ Vb,
                       const float* __restrict__ out, float* __restrict__ pred) {
  const int i = blockIdx.x;
  const float* h = out + DD;
  float acc = 0.0f;
  for (int c = threadIdx.x; c < H; c += 256) acc += Vw[i * H + c] * h[c];
  __shared__ float sred[8];
  for (int o = 16; o > 0; o >>= 1) acc += __shfl_down(acc, o);
  if ((threadIdx.x & 31) == 0) sred[threadIdx.x >> 5] = acc;
  __syncthreads();
  if (threadIdx.x == 0) {
    float t = 0.0f;
#pragma unroll
    for (int w = 0; w < 8; ++w) t += sred[w];
    pred[i] = t + Vb[i];
  }
}

// 10 segmented log_softmaxes over pred -> d_out[0..DD)
__global__ void k_lsm(const float* __restrict__ pred, float* __restrict__ out) {
  __shared__ float p[DD];
  if (threadIdx.x < DD) p[threadIdx.x] = pred[threadIdx.x];
  __syncthreads();
  const int seg[11] = {0, 13, 26, 39, 48, 52, 65, 78, 91, 100, 104};
  const int s = threadIdx.x;
  if (s < 10) {
    const int a = seg[s], b = seg[s + 1];
    float m = -3.0e38f;
    for (int k = a; k < b; ++k) m = fmaxf(m, p[k]);
    float sum = 0.0f;
    for (int k = a; k < b; ++k) sum += expf(p[k] - m);
    const float lse = m + logf(sum);
    for (int k = a; k < b; ++k) out[k] = p[k] - lse;
  }
}

extern "C" void kernel_launch(void* const* d_in, const int* in_sizes, int n_in,
                              void* d_out, int out_size, void* d_ws, size_t ws_size,
                              hipStream_t stream) {
  const float* line   = (const float*)d_in[0];
  const float* hidden = (const float*)d_in[1];
  const float* Uw     = (const float*)d_in[2];
  const float* Ub     = (const float*)d_in[3];
  const float* Ww     = (const float*)d_in[4];
  const float* Wb     = (const float*)d_in[5];
  const float* Vw     = (const float*)d_in[6];
  const float* Vb     = (const float*)d_in[7];
  float* out  = (float*)d_out;
  float* part = (float*)d_ws;            // KSPLIT * H floats
  float* pred = part + KSPLIT * H;       // DD floats

  dim3 g1(H / TILE_M, KSPLIT);           // 256 x 8 blocks
  k_gemv_wmma<<<g1, WAVES * 32, 0, stream>>>(Uw, hidden, Ww, line, part);
  k_finish_h<<<H / 256, 256, 0, stream>>>(part, Ub, Wb, out);
  k_pred<<<DD, 256, 0, stream>>>(Vw, Vb, out, pred);
  k_lsm<<<1, 128, 0, stream>>>(pred, out);
}